// RenderingNetwork_33303176413210
// MI455X (gfx1250) — compile-verified
//
#include <hip/hip_runtime.h>
#include <math.h>

// ---------------------------------------------------------------------------
// Problem constants (from reference)
// ---------------------------------------------------------------------------
#define NQ      16384          // R*S shading points
#define NP      4096           // physical particles
#define KNN     32             // ball_query K  (== wave32 width!)
#define FEAT    256
#define SRAYS   64             // S
#define RAD     0.1f           // 4.0 * 0.025
#define HID     512            // hidden width (din padded 511->512)

#define TILES_M 2              // 16-row tiles per wave  (32 rows)
#define TILES_N 4              // 16-col tiles per wave  (64 cols)

typedef float v2f __attribute__((ext_vector_type(2)));
typedef float v8f __attribute__((ext_vector_type(8)));

// ---------------------------------------------------------------------------
// Wave32 helpers
// ---------------------------------------------------------------------------
__device__ inline float wave_sum(float v) {
#pragma unroll
  for (int o = 16; o > 0; o >>= 1) v += __shfl_xor(v, o, 32);
  return v;
}

// ---------------------------------------------------------------------------
// Kernel 0: transpose weights to k-major Wt[k][n]; pads k >= din_src with 0
// (handles layer-0's 511 -> 512 padding)
// ---------------------------------------------------------------------------
__global__ void __launch_bounds__(256) transpose_w_kernel(
    const float* __restrict__ W, float* __restrict__ Wt, int din_src) {
  const int idx = blockIdx.x * blockDim.x + threadIdx.x;  // HID*HID threads
  const int k = idx >> 9;   // dest row (k index)
  const int n = idx & 511;  // dest col (output neuron)
  Wt[(size_t)k * HID + n] = (k < din_src) ? W[(size_t)n * din_src + k] : 0.0f;
}

// ---------------------------------------------------------------------------
// Kernel 1: ball query + neighborhood aggregates.  One wave per query point.
// Lane l ends up owning neighbor slot l (K == 32 == wave width).
// agg layout per query (12 floats): [density, sm.xyz, var.xyz, dir.xyz, 0, 0]
// ---------------------------------------------------------------------------
__global__ void __launch_bounds__(256) ballquery_agg_kernel(
    const float* __restrict__ pts,    // [NQ,3]
    const float* __restrict__ parts,  // [NP,3]
    const float* __restrict__ ro,     // [3]
    float* __restrict__ agg)          // [NQ,12]
{
  __shared__ float s_d2[8][32], s_nx[8][32], s_ny[8][32], s_nz[8][32];
  const int lane = threadIdx.x & 31;
  const int wv   = threadIdx.x >> 5;
  const int q    = blockIdx.x * 8 + wv;
  const bool active = (q < NQ);

  // zero-init neighbor slots (padding semantics of reference)
  s_d2[wv][lane] = 0.0f; s_nx[wv][lane] = 0.0f;
  s_ny[wv][lane] = 0.0f; s_nz[wv][lane] = 0.0f;

  float qx = 0.f, qy = 0.f, qz = 0.f;
  if (active) {
    qx = pts[q * 3 + 0]; qy = pts[q * 3 + 1]; qz = pts[q * 3 + 2];
    const float qq = qx * qx + qy * qy + qz * qz;
    const float r2 = RAD * RAD;
    int count = 0;
    for (int chunk = 0; chunk < NP / 32; ++chunk) {
      const int pi = chunk * 32 + lane;
      const float px = parts[pi * 3 + 0];
      const float py = parts[pi * 3 + 1];
      const float pz = parts[pi * 3 + 2];
      // reference-style d2 = |q|^2 + |p|^2 - 2 q.p, clamped at 0
      float d2 = qq + px * px + py * py + pz * pz
               - 2.0f * (qx * px + qy * py + qz * pz);
      d2 = fmaxf(d2, 0.0f);
      const bool inball = d2 < r2;
      const unsigned mask = (unsigned)__ballot(inball);
      if (inball) {
        const int slot = count + __popc(mask & ((1u << lane) - 1u));
        if (slot < KNN) {
          s_d2[wv][slot] = d2;
          s_nx[wv][slot] = px; s_ny[wv][slot] = py; s_nz[wv][slot] = pz;
        }
      }
      count += __popc(mask);
      if (count >= KNN) break;
    }
  }
  __syncthreads();
  if (!active) return;

  // lane owns slot `lane`
  const float d2s = s_d2[wv][lane];
  const float nx = s_nx[wv][lane], ny = s_ny[wv][lane], nz = s_nz[wv][lane];

  // smoothing weights use raw (possibly zero-padded) neighbor positions,
  // exactly like the reference
  const float ddx = nx - qx, ddy = ny - qy, ddz = nz - qz;
  const float d = sqrtf(ddx * ddx + ddy * ddy + ddz * ddz);
  const float t = d / RAD;
  const float w = fmaxf(1.0f - t * t * t, 0.0f);

  const float wsum = wave_sum(w);
  const float swx = wave_sum(w * nx);
  const float swy = wave_sum(w * ny);
  const float swz = wave_sum(w * nz);
  const float inv_ws = 1.0f / (wsum + 1e-12f);
  const float smx = swx * inv_ws, smy = swy * inv_ws, smz = swz * inv_ws;

  // masked variance (nn_mask = dists != 0, reference semantics)
  const float m = (d2s != 0.0f) ? 1.0f : 0.0f;
  const float num_nn = wave_sum(m);
  const float vx = m * ddx, vy = m * ddy, vz = m * ddz;
  const float inv_nn = 1.0f / (num_nn + 1e-12f);
  const float mx = wave_sum(vx) * inv_nn;
  const float my = wave_sum(vy) * inv_nn;
  const float mz = wave_sum(vz) * inv_nn;
  const float varx = wave_sum(m * (vx - mx) * (vx - mx)) * inv_nn;
  const float vary = wave_sum(m * (vy - my) * (vy - my)) * inv_nn;
  const float varz = wave_sum(m * (vz - mz) * (vz - mz)) * inv_nn;

  if (lane == 0) {
    float dx = smx - ro[0], dy = smy - ro[1], dz = smz - ro[2];
    const float inv_n = 1.0f / sqrtf(dx * dx + dy * dy + dz * dz);
    float* a = agg + (size_t)q * 12;
    a[0] = wsum;                     // density
    a[1] = smx;  a[2] = smy;  a[3] = smz;
    a[4] = varx; a[5] = vary; a[6] = varz;
    a[7] = dx * inv_n; a[8] = dy * inv_n; a[9] = dz * inv_n;
  }
}

// ---------------------------------------------------------------------------
// Kernel 2: build the 511-dim (padded to 512) MLP input row per query.
// ---------------------------------------------------------------------------
__device__ inline void embed_vec(float* __restrict__ row, int& c,
                                 const float* x, int C, int F) {
  for (int j = 0; j < C; ++j) row[c++] = x[j];
  float freq = 1.0f;
  for (int f = 0; f < F; ++f) {
    for (int j = 0; j < C; ++j) {
      row[c + j]     = sinf(freq * x[j]);
      row[c + C + j] = cosf(freq * x[j]);
    }
    c += 2 * C;
    freq *= 2.0f;
  }
}

__global__ void __launch_bounds__(256) build_features_kernel(
    const float* __restrict__ pts,     // [NQ,3]
    const float* __restrict__ normals, // [NQ,3]
    const float* __restrict__ feats,   // [NQ,FEAT]
    const float* __restrict__ rays,    // [R,6]
    const float* __restrict__ agg,     // [NQ,12]
    float* __restrict__ X)             // [NQ,HID]
{
  const int i = blockIdx.x * blockDim.x + threadIdx.x;
  if (i >= NQ) return;
  float* row = X + (size_t)i * HID;
  const float* a = agg + (size_t)i * 12;
  int c = 0;

  float q[3] = { pts[i * 3], pts[i * 3 + 1], pts[i * 3 + 2] };
  embed_vec(row, c, q, 3, 10);                       // 63
  float dens[1] = { a[0] };
  embed_vec(row, c, dens, 1, 4);                     // 9
  float sm[3] = { a[1], a[2], a[3] };
  embed_vec(row, c, sm, 3, 10);                      // 63
  float vr[3] = { a[4], a[5], a[6] };
  embed_vec(row, c, vr, 3, 10);                      // 63
  const int r = i / SRAYS;
  float hd[3] = { rays[r * 6 + 3], rays[r * 6 + 4], rays[r * 6 + 5] };
  embed_vec(row, c, hd, 3, 4);                       // 27
  float dir[3] = { a[7], a[8], a[9] };
  embed_vec(row, c, dir, 3, 4);                      // 27
  row[c++] = normals[i * 3 + 0];
  row[c++] = normals[i * 3 + 1];
  row[c++] = normals[i * 3 + 2];                     // -> 255 used
  const float* fv = feats + (size_t)i * FEAT;
  for (int j = 0; j < FEAT; ++j) row[c++] = fv[j];   // -> 511
  row[511] = 0.0f;                                   // pad column
}

// ---------------------------------------------------------------------------
// Kernel 3: f32 WMMA GEMM  Y[NQ,512] = act(X[NQ,512] @ Wt + b)
//   - one block = 32 rows x 512 cols; 8 waves, each wave 2x4 16x16 tiles
//   - A tile (32x512, 64 KB) staged in LDS, k-major with 16-row XOR swizzle
//     so both halves of every fragment read hit disjoint bank halves
//   - Wt is k-major => B fragment loads are coalesced b32 per half-wave
//   - V_WMMA_F32_16X16X4_F32, 8 independent accumulators per wave
// ---------------------------------------------------------------------------
__global__ void __launch_bounds__(256) gemm_wmma_f32_kernel(
    const float* __restrict__ X,    // [NQ, HID]
    const float* __restrict__ Wt,   // [HID, HID] k-major
    const float* __restrict__ bias, // [HID]
    float* __restrict__ Y,          // [NQ, HID]
    int relu)
{
  __shared__ float lds_a[32 * HID];          // exactly 64 KB
  const int t    = threadIdx.x;
  const int lane = t & 31;
  const int wv   = t >> 5;
  const int row0 = blockIdx.x * 32;

  // ---- stage A tile: element (r, k) -> lds_a[k*32 + (r ^ (((k>>1)&1)<<4))]
  {
    const int r   = t >> 3;                  // 0..31
    const int seg = t & 7;                   // 0..7 -> 64 consecutive cols
    const float* src = X + (size_t)(row0 + r) * HID + seg * 64;
#pragma unroll
    for (int c4 = 0; c4 < 16; ++c4) {
      const float4 v = *(const float4*)(src + c4 * 4);
      const int colb = seg * 64 + c4 * 4;
      const float vv[4] = { v.x, v.y, v.z, v.w };
#pragma unroll
      for (int u = 0; u < 4; ++u) {
        const int col = colb + u;
        const int rsw = r ^ (((col >> 1) & 1) << 4);
        lds_a[col * 32 + rsw] = vv[u];
      }
    }
  }
  __syncthreads();

  const int half    = lane >> 4;             // 0: k,k+1   1: k+2,k+3
  const int l15     = lane & 15;
  const int colbase = wv * (TILES_N * 16);   // 64-col group per wave
  // swizzled A rows for this lane (k%4==0 => swizzle bit == half)
  const int ar0 = l15 ^ (half << 4);
  const int ar1 = (l15 + 16) ^ (half << 4);

  v8f acc[TILES_M][TILES_N] = {};

#pragma unroll 2
  for (int k = 0; k < HID; k += 4) {
    const int kk = k + 2 * half;
    const v2f a0 = { lds_a[kk * 32 + ar0], lds_a[(kk + 1) * 32 + ar0] };
    const v2f a1 = { lds_a[kk * 32 + ar1], lds_a[(kk + 1) * 32 + ar1] };
    const float* wrow0 = Wt + (size_t)kk * HID + colbase + l15;
    const float* wrow1 = wrow0 + HID;
#pragma unroll
    for (int j = 0; j < TILES_N; ++j) {
      const v2f bj = { wrow0[j * 16], wrow1[j * 16] };
      acc[0][j] = __builtin_amdgcn_wmma_f32_16x16x4_f32(
          false, a0, false, bj, (short)0, acc[0][j], false, false);
      acc[1][j] = __builtin_amdgcn_wmma_f32_16x16x4_f32(
          false, a1, false, bj, (short)0, acc[1][j], false, false);
    }
  }

  // ---- epilogue: bias + optional relu, D layout M = v + 8*half, N = l15
#pragma unroll
  for (int j = 0; j < TILES_N; ++j) {
    const int col = colbase + j * 16 + l15;
    const float bb = bias[col];
#pragma unroll
    for (int i = 0; i < TILES_M; ++i) {
#pragma unroll
      for (int v = 0; v < 8; ++v) {
        float o = acc[i][j][v] + bb;
        if (relu) o = fmaxf(o, 0.0f);
        Y[(size_t)(row0 + i * 16 + v + 8 * half) * HID + col] = o;
      }
    }
  }
}

// ---------------------------------------------------------------------------
// Kernel 4: final layer 512 -> 3 + sigmoid
// ---------------------------------------------------------------------------
__global__ void __launch_bounds__(256) final_layer_kernel(
    const float* __restrict__ X, const float* __restrict__ W,
    const float* __restrict__ b, float* __restrict__ out)
{
  const int i = blockIdx.x * blockDim.x + threadIdx.x;
  if (i >= NQ) return;
  const float* x = X + (size_t)i * HID;
#pragma unroll
  for (int o = 0; o < 3; ++o) {
    const float* w = W + o * HID;
    float s = b[o];
    for (int k = 0; k < HID; ++k) s = fmaf(x[k], w[k], s);
    out[i * 3 + o] = 1.0f / (1.0f + expf(-s));
  }
}

// ---------------------------------------------------------------------------
// Host-side launcher
// ---------------------------------------------------------------------------
extern "C" void kernel_launch(void* const* d_in, const int* in_sizes, int n_in,
                              void* d_out, int out_size, void* d_ws, size_t ws_size,
                              hipStream_t stream) {
  const float* points  = (const float*)d_in[0];   // [R,S,3]
  const float* normals = (const float*)d_in[1];   // [N,3]
  // d_in[2] = view_dirs (unused by reference)
  const float* feats   = (const float*)d_in[3];   // [N,256]
  const float* parts   = (const float*)d_in[4];   // [P,3]
  const float* rays    = (const float*)d_in[5];   // [R,6]
  const float* ro      = (const float*)d_in[6];   // [3]
  const float* W0 = (const float*)d_in[7];
  const float* b0 = (const float*)d_in[8];
  const float* W1 = (const float*)d_in[9];
  const float* b1 = (const float*)d_in[10];
  const float* W2 = (const float*)d_in[11];
  const float* b2 = (const float*)d_in[12];
  const float* W3 = (const float*)d_in[13];
  const float* b3 = (const float*)d_in[14];
  const float* W4 = (const float*)d_in[15];
  const float* b4 = (const float*)d_in[16];

  // workspace layout (floats)
  float* bufA = (float*)d_ws;                         // [NQ, HID]
  float* bufB = bufA + (size_t)NQ * HID;              // [NQ, HID]
  float* Wt0  = bufB + (size_t)NQ * HID;              // [HID, HID] each
  float* Wt1  = Wt0 + (size_t)HID * HID;
  float* Wt2  = Wt1 + (size_t)HID * HID;
  float* Wt3  = Wt2 + (size_t)HID * HID;
  float* agg  = Wt3 + (size_t)HID * HID;              // [NQ, 12]

  // 0) transpose (and pad) hidden-layer weights to k-major
  transpose_w_kernel<<<(HID * HID) / 256, 256, 0, stream>>>(W0, Wt0, 511);
  transpose_w_kernel<<<(HID * HID) / 256, 256, 0, stream>>>(W1, Wt1, HID);
  transpose_w_kernel<<<(HID * HID) / 256, 256, 0, stream>>>(W2, Wt2, HID);
  transpose_w_kernel<<<(HID * HID) / 256, 256, 0, stream>>>(W3, Wt3, HID);

  // 1) ball query + aggregates: one wave per query, 8 waves per block
  ballquery_agg_kernel<<<NQ / 8, 256, 0, stream>>>(points, parts, ro, agg);

  // 2) feature rows
  build_features_kernel<<<NQ / 256, 256, 0, stream>>>(
      points, normals, feats, rays, agg, bufA);

  // 3) hidden MLP layers, WMMA f32, ping-pong bufA <-> bufB
  const int gblocks = NQ / 32;                        // 512 blocks, 8 waves ea.
  gemm_wmma_f32_kernel<<<gblocks, 256, 0, stream>>>(bufA, Wt0, b0, bufB, 1);
  gemm_wmma_f32_kernel<<<gblocks, 256, 0, stream>>>(bufB, Wt1, b1, bufA, 1);
  gemm_wmma_f32_kernel<<<gblocks, 256, 0, stream>>>(bufA, Wt2, b2, bufB, 1);
  gemm_wmma_f32_kernel<<<gblocks, 256, 0, stream>>>(bufB, Wt3, b3, bufA, 1);

  // 4) final 512->3 + sigmoid
  final_layer_kernel<<<NQ / 256, 256, 0, stream>>>(bufA, W4, b4, (float*)d_out);

  (void)in_sizes; (void)n_in; (void)out_size; (void)ws_size;
}